// VCReg_49520972923001
// MI455X (gfx1250) — compile-verified
//
#include <hip/hip_runtime.h>

// VCReg loss, y: (B=32, C=128, 48x48) f32, MI455X / gfx1250.
//
// cov term: per-sample hw x hw matrix collapsed algebraically onto the C x C
// Gram K = D D^T (D = y - mean_B):   ||M||^2 = (sumK)^2 - 2*sum_r rowsum_r^2 + ||K||^2
// K computed with V_WMMA_F32_16X16X4_F32 (wave32, full f32 precision).
// Split-K over hw (SPLIT=3 -> 96 WGs) for WGP occupancy; partial Grams are
// linear in K so they are atomically accumulated into a per-sample K buffer.
// LDS staging uses double-buffered GLOBAL_LOAD_ASYNC_TO_LDS_B128 (ASYNCcnt)
// when the builtin exists, else a plain float4 copy.

typedef __attribute__((ext_vector_type(2))) float v2f;
typedef __attribute__((ext_vector_type(8))) float v8f;

// Exact pointee type expected by the async-LDS builtins (per hipcc diagnostic:
// "__attribute__((__vector_size__(4 * sizeof(int)))) int __device__ *").
typedef int v4i_mem __attribute__((vector_size(16)));
typedef __attribute__((address_space(1))) v4i_mem* gas_v4p;   // global src
typedef __attribute__((address_space(3))) v4i_mem* las_v4p;   // LDS dst

#define B_N    32
#define C_N    128
#define HW_N   2304
#define CHW    (C_N * HW_N)          // 294912
#define SPLIT  3                     // K-segments per sample
#define SEG    (HW_N / SPLIT)        // 768
#define KC     64                    // K-chunk staged per LDS buffer
#define NCHUNK (SEG / KC)            // 12
#define DSTR   (KC + 4)              // padded LDS row stride (floats)

#if defined(__HIP_DEVICE_COMPILE__) && __has_builtin(__builtin_amdgcn_global_load_async_to_lds_b128)
#define USE_ASYNC 1
#else
#define USE_ASYNC 0
#endif

// workspace layout (floats):
//   [0..15]                       accumulators: [0]=std sum, [1]=cov sum
//   [WS_K  .. WS_K + 32*128*128)  per-sample Gram K (zeroed each call)
//   [WS_D  .. WS_D + B*C*HW)      D = y - mean_B
// total ~= 39.9 MB of d_ws.
#define WS_K   16
#define WS_D   (WS_K + B_N * C_N * C_N)      // 524304 (16B aligned)

__global__ void vcreg_init(float* ws, int n) {
    int i = blockIdx.x * blockDim.x + threadIdx.x;
    if (i < n) ws[i] = 0.0f;
}

// D[b][c][p] = y[b][c][p] - mean_b(y[.][c][p])  (one fused pass)
__global__ void vcreg_diff(const float* __restrict__ y, float* __restrict__ D) {
    int idx = blockIdx.x * blockDim.x + threadIdx.x;     // c*HW + p
    if (idx >= CHW) return;
    float v[B_N];
    float s = 0.0f;
#pragma unroll
    for (int b = 0; b < B_N; ++b) {
        v[b] = y[(size_t)b * CHW + idx];
        s += v[b];
    }
    float m = s * (1.0f / (float)B_N);
#pragma unroll
    for (int b = 0; b < B_N; ++b)
        D[(size_t)b * CHW + idx] = v[b] - m;
}

// std loss: flat (4096, 2304) view; unbiased var over 32 consecutive rows.
__global__ void vcreg_std(const float* __restrict__ y, float* __restrict__ acc) {
    int idx = blockIdx.x * blockDim.x + threadIdx.x;     // g*HW + p
    float contrib = 0.0f;
    if (idx < C_N * HW_N) {
        int g = idx / HW_N;
        int p = idx - g * HW_N;
        const float* base = y + (size_t)g * 32 * HW_N + p;
        float s = 0.0f, s2 = 0.0f;
#pragma unroll
        for (int j = 0; j < 32; ++j) {
            float v = base[(size_t)j * HW_N];
            s += v; s2 += v * v;
        }
        float var = (s2 - s * s * (1.0f / 32.0f)) * (1.0f / 31.0f);  // ddof=1
        float sd  = sqrtf(var + 1e-4f);
        contrib = fmaxf(1.0f - sd, 0.0f);
    }
#pragma unroll
    for (int m = 16; m >= 1; m >>= 1)
        contrib += __shfl_xor(contrib, m);
    if ((threadIdx.x & 31) == 0)
        atomicAdd(&acc[0], contrib);
}

// Stage one 128 x KC chunk of D into LDS. src = &D[sample][0][segBase + k0],
// row stride HW_N. 2048 x 16B transfers, 8 per thread, coalesced per row.
__device__ __forceinline__ void stage_chunk(const float* __restrict__ src,
                                            float* ldsBuf, int tid) {
#pragma unroll
    for (int j = 0; j < 8; ++j) {
        int u  = tid + 256 * j;          // 16-byte unit id, 16 units per row
        int c  = u >> 4;
        int k4 = (u & 15) << 2;          // float offset within the chunk row
        const float* g = src + (size_t)c * HW_N + k4;
        float* l = ldsBuf + c * DSTR + k4;
#if USE_ASYNC
        __builtin_amdgcn_global_load_async_to_lds_b128(
            (gas_v4p)(uintptr_t)g,
            (las_v4p)(uint32_t)(uintptr_t)l,
            /*offset=*/0, /*cpol=*/0);
#else
        *(float4*)l = *(const float4*)g;
#endif
    }
}

__device__ __forceinline__ void wait_async_le8() {
#if USE_ASYNC
#if __has_builtin(__builtin_amdgcn_s_wait_asynccnt)
    __builtin_amdgcn_s_wait_asynccnt(8);
#else
    asm volatile("s_wait_asynccnt 0x8" ::: "memory");
#endif
#endif
}
__device__ __forceinline__ void wait_async_le0() {
#if USE_ASYNC
#if __has_builtin(__builtin_amdgcn_s_wait_asynccnt)
    __builtin_amdgcn_s_wait_asynccnt(0);
#else
    asm volatile("s_wait_asynccnt 0x0" ::: "memory");
#endif
#endif
}

// One (sample, K-segment) per workgroup: partial Gram of 128x128 over 768 K,
// accumulated into the per-sample K buffer.
__global__ void __launch_bounds__(256) vcreg_cov(const float* __restrict__ D,
                                                 float* __restrict__ Kbuf) {
    __shared__ float ldsD[2][C_N * DSTR];    // double buffer, 2 x 34816 B

    const int tid  = threadIdx.x;
    const int wave = tid >> 5;               // ti: rows 16*wave .. 16*wave+15
    const int lane = tid & 31;
    const int half = lane >> 4;              // ISA 16x4 f32 A-layout K selector
    const int l16  = lane & 15;
    const int sample = blockIdx.x / SPLIT;
    const int seg    = blockIdx.x - sample * SPLIT;
    const float* base = D + (size_t)sample * CHW + seg * SEG;

    v8f accT[8] = {};                        // tiles (ti=wave, tj=0..7)

    stage_chunk(base, &ldsD[0][0], tid);     // prologue: chunk 0
    for (int n = 0; n < NCHUNK; ++n) {
        if (n + 1 < NCHUNK) {
            stage_chunk(base + (n + 1) * KC, &ldsD[(n + 1) & 1][0], tid);
            wait_async_le8();                // chunk n landed; n+1 in flight
        } else {
            wait_async_le0();
        }
        __syncthreads();                     // chunk n visible to all waves

        const float* buf   = &ldsD[n & 1][0];
        const float* Abase = &buf[(wave * 16 + l16) * DSTR + 2 * half];
        const float* Bbase[8];
#pragma unroll
        for (int tj = 0; tj < 8; ++tj)
            Bbase[tj] = &buf[(tj * 16 + l16) * DSTR + 2 * half];

        for (int kk = 0; kk < KC; kk += 4) {
            v2f a = *(const v2f*)(Abase + kk);
#pragma unroll
            for (int tj = 0; tj < 8; ++tj) {
                v2f b = *(const v2f*)(Bbase[tj] + kk);
                accT[tj] = __builtin_amdgcn_wmma_f32_16x16x4_f32(
                    /*neg_a=*/false, a, /*neg_b=*/false, b,
                    /*c_mod=*/(short)0, accT[tj],
                    /*reuse_a=*/false, /*reuse_b=*/false);
            }
        }
        __syncthreads();                     // buf[n&1] free before refill @ n+1
    }

    // Accumulate partial Gram into global per-sample K.
    // C/D layout: VGPR r, lanes 0-15 -> M=r, lanes 16-31 -> M=r+8; N=l16.
    float* Ks = Kbuf + (size_t)sample * (C_N * C_N);
#pragma unroll
    for (int tj = 0; tj < 8; ++tj)
#pragma unroll
        for (int r = 0; r < 8; ++r)
            atomicAdd(&Ks[(wave * 16 + r + 8 * half) * C_N + tj * 16 + l16],
                      accT[tj][r]);
}

// Per sample: reduce K -> c_i. S1=sumK, S2=sum_r rowsum^2, S3=||K||^2.
__global__ void __launch_bounds__(256) vcreg_reduce(const float* __restrict__ Kbuf,
                                                    float* __restrict__ acc) {
    __shared__ float ldsScal[3];
    const int tid = threadIdx.x;
    if (tid < 3) ldsScal[tid] = 0.0f;
    __syncthreads();

    const float* Ks = Kbuf + (size_t)blockIdx.x * (C_N * C_N);
    float rsum = 0.0f, s3 = 0.0f;
    if (tid < C_N) {
        const float* row = Ks + tid * C_N;
        for (int n = 0; n < C_N; ++n) {
            float v = row[n];
            rsum += v; s3 += v * v;
        }
    }
    float s1 = rsum, s2 = rsum * rsum;
#pragma unroll
    for (int m = 16; m >= 1; m >>= 1) {
        s1 += __shfl_xor(s1, m);
        s2 += __shfl_xor(s2, m);
        s3 += __shfl_xor(s3, m);
    }
    if ((tid & 31) == 0 && tid < C_N) {
        atomicAdd(&ldsScal[0], s1);
        atomicAdd(&ldsScal[1], s2);
        atomicAdd(&ldsScal[2], s3);
    }
    __syncthreads();

    if (tid == 0) {
        double S1 = (double)ldsScal[0];
        double S2 = (double)ldsScal[1];
        double S3 = (double)ldsScal[2];
        double M2 = S1 * S1 - 2.0 * S2 + S3;     // ||outer(s,s) - d^T d||_F^2
        double denom = (double)HW_N * (double)(C_N - 1);
        atomicAdd(&acc[1], (float)(M2 / (denom * denom) / (double)HW_N));
    }
}

__global__ void vcreg_final(const float* __restrict__ acc, float* __restrict__ out) {
    out[0] = 25.0f * acc[0] / (float)(C_N * HW_N) + acc[1] / (float)B_N;
}

extern "C" void kernel_launch(void* const* d_in, const int* in_sizes, int n_in,
                              void* d_out, int out_size, void* d_ws, size_t ws_size,
                              hipStream_t stream) {
    (void)in_sizes; (void)n_in; (void)out_size; (void)ws_size;
    const float* y = (const float*)d_in[0];
    float* out  = (float*)d_out;
    float* wsf  = (float*)d_ws;
    float* accs = wsf;                 // [0]=std sum, [1]=cov sum
    float* Kbuf = wsf + WS_K;          // 32 x 128 x 128
    float* Dbuf = wsf + WS_D;          // B x C x HW

    const int n  = CHW;                // 294912
    const int nz = WS_D;               // zero accs + Kbuf

    vcreg_init  <<<(nz + 255) / 256, 256, 0, stream>>>(wsf, nz);
    vcreg_diff  <<<(n + 255) / 256, 256, 0, stream>>>(y, Dbuf);
    vcreg_std   <<<(n + 255) / 256, 256, 0, stream>>>(y, accs);
    vcreg_cov   <<<B_N * SPLIT, 256, 0, stream>>>(Dbuf, Kbuf);
    vcreg_reduce<<<B_N, 256, 0, stream>>>(Kbuf, accs);
    vcreg_final <<<1, 1, 0, stream>>>(accs, out);
}